// PatternLoss_2D_44152263803103
// MI455X (gfx1250) — compile-verified
//
#include <hip/hip_runtime.h>

typedef __attribute__((ext_vector_type(4))) float v4f;
typedef __attribute__((ext_vector_type(2))) float v2f;
typedef __attribute__((ext_vector_type(8))) float v8f;

constexpr int W      = 512;
constexpr int H      = 512;
constexpr int NIMG   = 64;
constexpr int CHUNKS = 16;   // 16 chunks x 32 columns = 512
constexpr int STRIPS = 34;   // 34 strips x 15 rows = 510 output rows
constexpr int SROWS  = 15;
constexpr int NBINS  = 512;
constexpr int THREADS_PER_TENSOR = NIMG * CHUNKS * STRIPS;   // 34816
constexpr int BLOCKS_PER_TENSOR  = THREADS_PER_TENSOR / 256; // 136

// Pack 34 binary columns (img > 0) of one row into a REVERSED mask:
// bit j of m == binarized pixel at column (c0 + 33 - j).
// Then row-code for output column c is (m >> (31-c)) & 7, already in
// MSB-first order (weights 4,2,1 over columns c,c+1,c+2).
__device__ __forceinline__ unsigned long long load_mask34(const float* rowp, bool extra) {
    unsigned long long m = 0ull;
#pragma unroll
    for (int q = 0; q < 8; ++q) {
        v4f v = __builtin_nontemporal_load((const v4f*)rowp + q);  // streamed once -> NT
        const int i = q * 4;
        if (v.x > 0.0f) m |= (1ull << (33 - i));
        if (v.y > 0.0f) m |= (1ull << (32 - i));
        if (v.z > 0.0f) m |= (1ull << (31 - i));
        if (v.w > 0.0f) m |= (1ull << (30 - i));
    }
    if (extra) {  // columns c0+32, c0+33 (not present for the last chunk)
        v2f e = __builtin_nontemporal_load((const v2f*)(rowp + 32));
        if (e.x > 0.0f) m |= 2ull;  // bit 1
        if (e.y > 0.0f) m |= 1ull;  // bit 0
    }
    return m;
}

__global__ void pattern_hist_kernel(const float* __restrict__ input,
                                    const float* __restrict__ target,
                                    unsigned* __restrict__ ghist) {
    __shared__ unsigned lhist[NBINS];
    for (int i = threadIdx.x; i < NBINS; i += blockDim.x) lhist[i] = 0u;
    __syncthreads();

    const int tensor = blockIdx.y;
    const float* src = tensor ? target : input;
    unsigned* gh = ghist + tensor * NBINS;

    const int tid   = blockIdx.x * blockDim.x + threadIdx.x;  // exact fit, no guard
    const int strip = tid % STRIPS;
    const int rest  = tid / STRIPS;
    const int chunk = rest % CHUNKS;
    const int img   = rest / CHUNKS;

    const int  c0    = chunk * 32;
    const bool extra = (chunk != CHUNKS - 1);
    const int  ncols = extra ? 32 : 30;      // last chunk: output cols 480..509
    const int  r0    = strip * SROWS;
    const float* base = src + (size_t)img * (W * H) + c0;

    // rolling 3-row window of reversed bitmasks
    unsigned long long m0 = load_mask34(base + (size_t)(r0 + 0) * W, extra);
    unsigned long long m1 = load_mask34(base + (size_t)(r0 + 1) * W, extra);

    for (int r = 0; r < SROWS; ++r) {
        unsigned long long m2 = load_mask34(base + (size_t)(r0 + r + 2) * W, extra);
#pragma unroll 8
        for (int c = 0; c < ncols; ++c) {
            const int s = 31 - c;
            const unsigned code =
                (((unsigned)(m0 >> s) & 7u) << 6) |
                (((unsigned)(m1 >> s) & 7u) << 3) |
                ( (unsigned)(m2 >> s) & 7u);
            atomicAdd(&lhist[code], 1u);          // ds_add_u32
        }
        m0 = m1; m1 = m2;
    }

    __syncthreads();
    for (int i = threadIdx.x; i < NBINS; i += blockDim.x)
        atomicAdd(&gh[i], lhist[i]);              // 512 global atomics per block
}

__global__ void zero_hist_kernel(unsigned* __restrict__ g) {
    g[blockIdx.x * blockDim.x + threadIdx.x] = 0u;
}

// Final reduction: loss = sum((h_in - h_tgt)^2) / 512 / 64, done on one full
// wave with V_WMMA_F32_16X16X4_F32:  D = A(e values, 16x4) x B(ones) + C.
// After 8 accumulations C[m][n] = sum of all 512 squared diffs split over M;
// fold 8 accumulator VGPRs per lane + lane^16 partner.
__global__ void pattern_mse_kernel(const unsigned* __restrict__ ghist,
                                   float* __restrict__ out) {
    const int l = threadIdx.x;                     // 32 threads, EXEC all ones
    const float inv_n = 1.0f / (float)(NIMG * 510 * 510);

    v8f acc = {};
    v2f b; b.x = 1.0f; b.y = 1.0f;                 // ones matrix (layout-agnostic)
    // 32-bit A 16x4 layout: lanes 0-15 hold {K=0,K=1}, lanes 16-31 hold {K=2,K=3}
    const int base = (l & 15) * 4 + (l >> 4) * 2;  // A[m][k] = e[t*64 + m*4 + k]
#pragma unroll
    for (int t = 0; t < 8; ++t) {
        const int i0 = t * 64 + base;
        const float d0 = ((float)ghist[i0]     - (float)ghist[NBINS + i0])     * inv_n;
        const float d1 = ((float)ghist[i0 + 1] - (float)ghist[NBINS + i0 + 1]) * inv_n;
        v2f a; a.x = d0 * d0; a.y = d1 * d1;
        acc = __builtin_amdgcn_wmma_f32_16x16x4_f32(
            /*neg_a=*/false, a, /*neg_b=*/false, b,
            /*c_mod=*/(short)0, acc, /*reuse_a=*/false, /*reuse_b=*/false);
    }
    float s = acc[0] + acc[1] + acc[2] + acc[3] + acc[4] + acc[5] + acc[6] + acc[7];
    s += __shfl_xor(s, 16, 32);                    // M=0..7 (+) M=8..15
    if (l == 0) out[0] = s * (1.0f / (float)NBINS) * (1.0f / (float)NIMG);
}

extern "C" void kernel_launch(void* const* d_in, const int* in_sizes, int n_in,
                              void* d_out, int out_size, void* d_ws, size_t ws_size,
                              hipStream_t stream) {
    (void)in_sizes; (void)n_in; (void)out_size; (void)ws_size;
    const float* input  = (const float*)d_in[0];
    const float* target = (const float*)d_in[1];
    unsigned* ghist = (unsigned*)d_ws;             // 2 x 512 u32 = 4 KB scratch

    zero_hist_kernel<<<1, 2 * NBINS, 0, stream>>>(ghist);
    pattern_hist_kernel<<<dim3(BLOCKS_PER_TENSOR, 2), 256, 0, stream>>>(input, target, ghist);
    pattern_mse_kernel<<<1, 32, 0, stream>>>(ghist, (float*)d_out);
}